// WeightedConv_24386824306930
// MI455X (gfx1250) — compile-verified
//
#include <hip/hip_runtime.h>

typedef float v2f __attribute__((ext_vector_type(2)));
typedef float v8f __attribute__((ext_vector_type(8)));

#define HID 32

// out[i] = 0.5f * feature[i]  (seeds the residual term; edge kernel atomically adds 0.5*message)
__global__ void wconv_seed_kernel(const float* __restrict__ feature,
                                  float* __restrict__ out, int n4) {
    int i = blockIdx.x * blockDim.x + threadIdx.x;
    if (i < n4) {
        const float4* f4 = (const float4*)feature;
        float4* o4 = (float4*)out;
        float4 v = f4[i];
        v.x *= 0.5f; v.y *= 0.5f; v.z *= 0.5f; v.w *= 0.5f;
        o4[i] = v;
    }
}

// One wave32 per edge. message = f(1x32) * W(32x32) via 16x V_WMMA_F32_16X16X4_F32.
//   A(16x4): f[4c..4c+3] broadcast to all 16 rows (lanes 0-15 hold K=0,1; 16-31 hold K=2,3)
//   B(4x16): W rows 4c..4c+3, col block nb*16..+15 (VGPR p: lo lanes K=p, hi lanes K=p+2)
//   D: all 16 rows identical == partial message.
__global__ void __launch_bounds__(256)
wconv_edge_kernel(const float* __restrict__ feature,
                  const float* __restrict__ weight,
                  const int* __restrict__ src,
                  const int* __restrict__ dst,
                  float* __restrict__ out,
                  int n_edges) {
    const int lane   = threadIdx.x & 31;
    const int wave   = threadIdx.x >> 5;
    const int edge   = blockIdx.x * (blockDim.x >> 5) + wave;
    if (edge >= n_edges) return;   // wave-uniform: EXEC stays all-ones inside

    const int s = src[edge];
    const int d = dst[edge];

    // Gather one feature row, lane-striped: f held by lane j is feature[s][j].
    const float f = feature[s * HID + lane];

    const float* __restrict__ w = weight + (size_t)edge * (HID * HID);

    const int hi     = lane >> 4;   // 0: lanes 0-15, 1: lanes 16-31
    const int lane15 = lane & 15;

    v8f acc0 = {0.f,0.f,0.f,0.f,0.f,0.f,0.f,0.f};  // cols  0..15
    v8f acc1 = {0.f,0.f,0.f,0.f,0.f,0.f,0.f,0.f};  // cols 16..31

#pragma unroll
    for (int c = 0; c < 8; ++c) {
        const int kbase = 4 * c + 2 * hi;

        // A fragment: every M-row = f chunk. Lo half wants k=4c,4c+1; hi half k=4c+2,4c+3.
        v2f a;
        a.x = __shfl(f, kbase,     32);
        a.y = __shfl(f, kbase + 1, 32);

        // B fragments: VGPR0 -> row kbase, VGPR1 -> row kbase+1 (per half-wave).
        // Non-temporal: 1 GiB of weights streams once, keep it out of L2.
        const int r0 = kbase * HID;          // row kbase   base offset
        const int r1 = r0 + HID;             // row kbase+1 base offset
        v2f b0, b1;
        b0.x = __builtin_nontemporal_load(&w[r0 + lane15]);
        b0.y = __builtin_nontemporal_load(&w[r1 + lane15]);
        b1.x = __builtin_nontemporal_load(&w[r0 + 16 + lane15]);
        b1.y = __builtin_nontemporal_load(&w[r1 + 16 + lane15]);

        acc0 = __builtin_amdgcn_wmma_f32_16x16x4_f32(false, a, false, b0,
                                                     (short)0, acc0, false, false);
        acc1 = __builtin_amdgcn_wmma_f32_16x16x4_f32(false, a, false, b1,
                                                     (short)0, acc1, false, false);
    }

    // D rows all identical: lane n<16 reads row M=0 (VGPR0, cols 0-15) of acc0;
    // lanes 16-31 read row M=8 copy (VGPR0, cols 16-31) of acc1.
    const float msg = hi ? acc1[0] : acc0[0];

    // Scatter-sum with the 0.5 residual factor folded in.
    atomicAdd(&out[d * HID + lane], 0.5f * msg);
}

extern "C" void kernel_launch(void* const* d_in, const int* in_sizes, int n_in,
                              void* d_out, int out_size, void* d_ws, size_t ws_size,
                              hipStream_t stream) {
    const float* feature = (const float*)d_in[0];
    const float* weight  = (const float*)d_in[1];
    const int*   src     = (const int*)d_in[2];
    const int*   dst     = (const int*)d_in[3];
    float*       out     = (float*)d_out;

    const int n_feat  = in_sizes[0];   // n_nodes * 32
    const int n_edges = in_sizes[2];

    // 1) out = 0.5 * feature
    const int n4 = n_feat / 4;
    wconv_seed_kernel<<<(n4 + 255) / 256, 256, 0, stream>>>(feature, out, n4);

    // 2) per-edge WMMA matvec + atomic scatter of 0.5*message
    const int waves_per_block = 8;     // 256 threads
    const int blocks = (n_edges + waves_per_block - 1) / waves_per_block;
    wconv_edge_kernel<<<blocks, 256, 0, stream>>>(feature, weight, src, dst, out, n_edges);
}